// PureSSM_76098230550864
// MI455X (gfx1250) — compile-verified
//
#include <hip/hip_runtime.h>
#include <hip/hip_bf16.h>
#include <math.h>

// Problem dims (fixed by the reference)
#define D_MODEL_   1024
#define D_STATE_   16
#define D_CONV_    4
#define D_INNER_   2048
#define BATCH_     4
#define SEQ_       2048
#define NTOK_      (BATCH_ * SEQ_)     // 8192 tokens
#define XZ_COLS_   (2 * D_INNER_)      // 4096

typedef float v2f __attribute__((ext_vector_type(2)));
typedef float v8f __attribute__((ext_vector_type(8)));
typedef int   v4i __attribute__((vector_size(16)));    // matches builtin param

#define GLOBAL_AS __attribute__((address_space(1)))
#define LDS_AS    __attribute__((address_space(3)))

#if __has_builtin(__builtin_amdgcn_global_load_async_to_lds_b128)
#define HAS_ASYNC_LDS 1
#endif

__device__ __forceinline__ void async_copy_b128(const float* g, float* l) {
#ifdef HAS_ASYNC_LDS
    __builtin_amdgcn_global_load_async_to_lds_b128(
        (GLOBAL_AS v4i*)g, (LDS_AS v4i*)l, 0, 0);
#endif
}

__device__ __forceinline__ void wait_async_lds() {
#ifdef HAS_ASYNC_LDS
#if __has_builtin(__builtin_amdgcn_s_wait_asynccnt)
    __builtin_amdgcn_s_wait_asynccnt(0);
#else
    asm volatile("s_wait_asynccnt 0x0" ::: "memory");
#endif
#endif
}

// ---------------------------------------------------------------------------
// GEMM: C[M,N] = A[M,K] (row stride lda)  x  W[N,K]^T  via V_WMMA_F32_16X16X4_F32
// Block tile 128x64, K-step 16, 8 waves (each owns 32x32 = 2x2 WMMA tiles).
// Double-buffered LDS; tiles staged with GLOBAL_LOAD_ASYNC_TO_LDS_B128 (DMA to
// LDS, ASYNCcnt) so the next K-step streams in while WMMAs run on the current.
// Both A and B tiles are stored row-major [row][k] with padded stride 20
// (conflict-free fragment reads, 16B-aligned async chunks, no transpose).
// ---------------------------------------------------------------------------
#define BM 128
#define BN 64
#define BK 16
#define BKP 20   // padded LDS row stride (floats): 80B rows, 16B-aligned chunks

__global__ __launch_bounds__(256)
void gemm_wmma_f32(const float* __restrict__ A, const float* __restrict__ W,
                   float* __restrict__ C, int M, int N, int K, int lda, int ldc)
{
    __shared__ float As[2][BM * BKP];   // As[buf][m*BKP + k]
    __shared__ float Bs[2][BN * BKP];   // Bs[buf][n*BKP + k]

    const int tid   = threadIdx.x;
    const int lane  = tid & 31;
    const int wid   = tid >> 5;
    const int waveM = wid & 3;       // 4 wave-rows of 32
    const int waveN = wid >> 2;      // 2 wave-cols of 32

    const int m0 = blockIdx.y * BM;
    const int n0 = blockIdx.x * BN;

    const int lr    = lane & 15;          // position within 16-lane half
    const int khi   = (lane >> 4) << 1;   // 0 or 2: K offset for A/B frags
    const int rowhi = (lane >> 4) << 3;   // 0 or 8: M offset for C/D store

    // per-thread staging coordinates
    const int ar = tid >> 1;              // A: row 0..127
    const int ac = (tid & 1) << 3;        //    k-offset 0 or 8 (2 x 16B)
    const int bn = tid >> 2;              // B: row (n) 0..63
    const int bq = (tid & 3) << 2;        //    k-offset 0,4,8,12 (1 x 16B)

    v8f acc[2][2] = {};

    // ---- tile stage: async DMA global -> LDS (fallback: sync copy) ----
    auto stage = [&](int buf, int k0) {
#ifdef HAS_ASYNC_LDS
        const float* asrc = A + (size_t)(m0 + ar) * lda + k0 + ac;
        async_copy_b128(asrc,     &As[buf][ar * BKP + ac]);
        async_copy_b128(asrc + 4, &As[buf][ar * BKP + ac + 4]);
        const float* bsrc = W + (size_t)(n0 + bn) * K + k0 + bq;
        async_copy_b128(bsrc,     &Bs[buf][bn * BKP + bq]);
#else
        const float* asrc = A + (size_t)(m0 + ar) * lda + k0 + ac;
        float4 va0 = *(const float4*)(asrc);
        float4 va1 = *(const float4*)(asrc + 4);
        *(float4*)&As[buf][ar * BKP + ac]     = va0;
        *(float4*)&As[buf][ar * BKP + ac + 4] = va1;
        const float* bsrc = W + (size_t)(n0 + bn) * K + k0 + bq;
        *(float4*)&Bs[buf][bn * BKP + bq] = *(const float4*)(bsrc);
#endif
    };

    stage(0, 0);

    int buf = 0;
    for (int k0 = 0; k0 < K; k0 += BK, buf ^= 1) {
        wait_async_lds();        // my async writes to LDS have landed
        __syncthreads();         // everyone's have; prev compute done reading

        if (k0 + BK < K)
            stage(buf ^ 1, k0 + BK);   // prefetch next tile while computing

        #pragma unroll
        for (int kk = 0; kk < BK; kk += 4) {
            // A frag (16x4 f32): lanes 0-15 hold K=kk,kk+1; lanes 16-31 kk+2,kk+3
            v2f afrag[2], bfrag[2];
            #pragma unroll
            for (int ti = 0; ti < 2; ++ti) {
                int m = waveM * 32 + ti * 16 + lr;
                afrag[ti].x = As[buf][m * BKP + kk + khi];
                afrag[ti].y = As[buf][m * BKP + kk + khi + 1];
            }
            // B frag (4x16 f32): lanes 0-15 rows kk,kk+1; lanes 16-31 kk+2,kk+3
            #pragma unroll
            for (int tj = 0; tj < 2; ++tj) {
                int n = waveN * 32 + tj * 16 + lr;
                bfrag[tj].x = Bs[buf][n * BKP + kk + khi];
                bfrag[tj].y = Bs[buf][n * BKP + kk + khi + 1];
            }
            #pragma unroll
            for (int ti = 0; ti < 2; ++ti)
                #pragma unroll
                for (int tj = 0; tj < 2; ++tj)
                    acc[ti][tj] = __builtin_amdgcn_wmma_f32_16x16x4_f32(
                        false, afrag[ti], false, bfrag[tj],
                        (short)0, acc[ti][tj], false, false);
        }
    }

    // ---- store: D layout = VGPR r -> (M = r [+8 for hi half], N = lane&15) ----
    #pragma unroll
    for (int ti = 0; ti < 2; ++ti) {
        int mbase = m0 + waveM * 32 + ti * 16 + rowhi;
        #pragma unroll
        for (int tj = 0; tj < 2; ++tj) {
            int n = n0 + waveN * 32 + tj * 16 + lr;
            #pragma unroll
            for (int r = 0; r < 8; ++r)
                C[(size_t)(mbase + r) * ldc + n] = acc[ti][tj][r];
        }
    }
}

// ---------------------------------------------------------------------------
// Depthwise causal conv (4 taps over seq) + SiLU. Reads x_proj half of xz.
// ---------------------------------------------------------------------------
__global__ __launch_bounds__(256)
void conv_silu_kernel(const float* __restrict__ xz,
                      const float* __restrict__ conv_w,
                      const float* __restrict__ conv_b,
                      float* __restrict__ xc)
{
    int idx = blockIdx.x * blockDim.x + threadIdx.x;   // over NTOK_*D_INNER_
    int d   = idx & (D_INNER_ - 1);
    int tok = idx >> 11;                               // D_INNER_ = 2048
    int l   = tok & (SEQ_ - 1);

    float acc = conv_b[d];
    #pragma unroll
    for (int k = 0; k < D_CONV_; ++k) {
        int ls = l + k - (D_CONV_ - 1);
        if (ls >= 0)
            acc += xz[(size_t)(tok + k - (D_CONV_ - 1)) * XZ_COLS_ + d]
                   * conv_w[d * D_CONV_ + k];
    }
    xc[idx] = acc / (1.f + __expf(-acc));              // SiLU
}

// ---------------------------------------------------------------------------
// x_dbl = x_conv @ W_xproj^T (N=33: too skinny for WMMA). One block per token;
// row cached in LDS, 4-way split-K per output, LDS reduce. Produces B, C, dt.
// ---------------------------------------------------------------------------
__global__ __launch_bounds__(256)
void xproj_kernel(const float* __restrict__ xc, const float* __restrict__ Wx,
                  float* __restrict__ Bm, float* __restrict__ Cm,
                  float* __restrict__ dtv)
{
    __shared__ float xs[D_INNER_];
    __shared__ float part[33 * 4];
    const int tok = blockIdx.x;
    const int tid = threadIdx.x;

    for (int k = tid; k < D_INNER_; k += 256)
        xs[k] = xc[(size_t)tok * D_INNER_ + k];
    __syncthreads();

    if (tid < 33 * 4) {
        int j = tid >> 2, p = tid & 3;
        const float* w = Wx + (size_t)j * D_INNER_;
        float sum = 0.f;
        for (int k = p * 512; k < (p + 1) * 512; ++k)
            sum = fmaf(xs[k], w[k], sum);
        part[tid] = sum;
    }
    __syncthreads();

    if (tid < 33) {
        float s = part[tid * 4] + part[tid * 4 + 1]
                + part[tid * 4 + 2] + part[tid * 4 + 3];
        if (tid < 16)      Bm[tok * D_STATE_ + tid] = s;
        else if (tid < 32) Cm[tok * D_STATE_ + (tid - 16)] = s;
        else {
            float sp = (s > 20.f) ? s : log1pf(__expf(s));   // softplus
            dtv[tok] = fminf(fmaxf(sp, 0.001f), 0.1f);
        }
    }
}

// ---------------------------------------------------------------------------
// Selective scan: one lane per (channel, state); 2048 serial steps.
// y reduction over 16 states via shfl_xor tree. Fuses the SiLU(z) gate +
// D skip and writes gated y into the dead x_proj slots of xz (lda=4096 for
// the output GEMM). Prefetches the streaming x_conv reads.
// ---------------------------------------------------------------------------
__global__ __launch_bounds__(256)
void scan_kernel(const float* __restrict__ xc, float* __restrict__ xz,
                 const float* __restrict__ Bm, const float* __restrict__ Cm,
                 const float* __restrict__ dtv, const float* __restrict__ A_log,
                 const float* __restrict__ D_param)
{
    int gid = blockIdx.x * blockDim.x + threadIdx.x;
    int ch  = gid >> 4;            // 0..8191 (b * D_INNER_ + d)
    int s   = gid & 15;            // state index
    int b   = ch >> 11;
    int d   = ch & (D_INNER_ - 1);

    const float A_s = -__expf(A_log[s]);   // A = -exp(A_log)
    const float Dp  = D_param[d];
    float h = 0.f;

    int tok = b * SEQ_;
    for (int l = 0; l < SEQ_; ++l, ++tok) {
        float xt  = xc[(size_t)tok * D_INNER_ + d];
        float dtt = dtv[tok];
        float Bv  = Bm[tok * D_STATE_ + s];
        float Cv  = Cm[tok * D_STATE_ + s];
        if (l + 16 < SEQ_)
            __builtin_prefetch(&xc[(size_t)(tok + 16) * D_INNER_ + d], 0, 0);

        float dtA  = fminf(fmaxf(dtt * A_s, -20.f), 0.f);
        float Abar = __expf(dtA);
        float Bbar = fminf(fmaxf(dtt * Bv, -10.f), 10.f);
        h = fminf(fmaxf(Abar * h + Bbar * xt, -100.f), 100.f);

        float contrib = h * Cv;                 // einsum over states
        contrib += __shfl_xor(contrib, 1);
        contrib += __shfl_xor(contrib, 2);
        contrib += __shfl_xor(contrib, 4);
        contrib += __shfl_xor(contrib, 8);

        if (s == 0) {
            float zv = xz[(size_t)tok * XZ_COLS_ + D_INNER_ + d];
            float sz = zv / (1.f + __expf(-zv));          // SiLU(z)
            xz[(size_t)tok * XZ_COLS_ + d] = contrib * sz + xt * Dp;
        }
    }
}

// ---------------------------------------------------------------------------
// Launch. Workspace layout (floats):
//   xz  : NTOK_*4096  (134.2 MB)  -- x_proj half is recycled for gated y
//   xc  : NTOK_*2048  ( 67.1 MB)
//   Bm  : NTOK_*16, Cm: NTOK_*16, dt: NTOK_   (~1.1 MB)
// total ~203 MB of d_ws.
// ---------------------------------------------------------------------------
extern "C" void kernel_launch(void* const* d_in, const int* in_sizes, int n_in,
                              void* d_out, int out_size, void* d_ws, size_t ws_size,
                              hipStream_t stream)
{
    (void)in_sizes; (void)n_in; (void)out_size; (void)ws_size;

    const float* x      = (const float*)d_in[0];
    const float* W_in   = (const float*)d_in[1];
    const float* conv_w = (const float*)d_in[2];
    const float* conv_b = (const float*)d_in[3];
    const float* W_x    = (const float*)d_in[4];
    const float* A_log  = (const float*)d_in[5];
    const float* D_par  = (const float*)d_in[6];
    const float* W_out  = (const float*)d_in[7];
    float* out = (float*)d_out;

    float* xz  = (float*)d_ws;
    float* xc  = xz + (size_t)NTOK_ * XZ_COLS_;
    float* Bm  = xc + (size_t)NTOK_ * D_INNER_;
    float* Cm  = Bm + (size_t)NTOK_ * D_STATE_;
    float* dtv = Cm + (size_t)NTOK_ * D_STATE_;

    // 1) xz = x @ W_in^T          (8192x1024)x(1024x4096)
    gemm_wmma_f32<<<dim3(XZ_COLS_ / BN, NTOK_ / BM), 256, 0, stream>>>(
        x, W_in, xz, NTOK_, XZ_COLS_, D_MODEL_, D_MODEL_, XZ_COLS_);

    // 2) depthwise conv + SiLU
    conv_silu_kernel<<<(NTOK_ * D_INNER_) / 256, 256, 0, stream>>>(
        xz, conv_w, conv_b, xc);

    // 3) x_dbl -> B, C, dt
    xproj_kernel<<<NTOK_, 256, 0, stream>>>(xc, W_x, Bm, Cm, dtv);

    // 4) selective scan + gate (writes y into xz[:, :D_INNER_])
    scan_kernel<<<(NTOK_ * D_STATE_) / 256, 256, 0, stream>>>(
        xc, xz, Bm, Cm, dtv, A_log, D_par);

    // 5) out = y @ W_out^T        (8192x2048)x(2048x1024), A row stride 4096
    gemm_wmma_f32<<<dim3(D_MODEL_ / BN, NTOK_ / BM), 256, 0, stream>>>(
        xz, W_out, out, NTOK_, D_MODEL_, D_INNER_, XZ_COLS_, D_MODEL_);
}